// ChamferDistancePy_41540923687002
// MI455X (gfx1250) — compile-verified
//
#include <hip/hip_runtime.h>
#include <math.h>

typedef float v2f __attribute__((ext_vector_type(2)));
typedef float v8f __attribute__((ext_vector_type(8)));

#define NPTS 16384
#define CSPLIT 2                       // column-sweep split (gridDim.y)
#define COLS_PER (NPTS / CSPLIT)       // 8192 columns per block
#define TILES_PER (COLS_PER / 16)      // 512 column tiles per block
#define INF_BITS 0x7F800000u

// ---------------------------------------------------------------------------
// Kernel 1: repack points into WMMA-friendly 16B rows, init global min tables.
//   p1pack[i] = ( x,  y,  z,  1 )             (A rows, K padded to 4)
//   sq1arr[i] = ||p1_i||^2                    (C fragment values)
//   p2pack[j] = (-2x, -2y, -2z, ||p2_j||^2)   (B cols: A.B = -2 a.b + ||b||^2)
// ---------------------------------------------------------------------------
__global__ void chamfer_prep_kernel(const float* __restrict__ p1,
                                    const float* __restrict__ p2,
                                    float4* __restrict__ p1pack,
                                    float4* __restrict__ p2pack,
                                    float* __restrict__ sq1arr,
                                    unsigned* __restrict__ rowmin_bits,
                                    unsigned* __restrict__ colmin_bits) {
    int i = blockIdx.x * blockDim.x + threadIdx.x;
    if (i >= NPTS) return;
    float x1 = p1[3 * i + 0], y1 = p1[3 * i + 1], z1 = p1[3 * i + 2];
    float s1 = x1 * x1 + y1 * y1 + z1 * z1;
    p1pack[i] = make_float4(x1, y1, z1, 1.0f);
    sq1arr[i] = s1;
    float x2 = p2[3 * i + 0], y2 = p2[3 * i + 1], z2 = p2[3 * i + 2];
    float s2 = x2 * x2 + y2 * y2 + z2 * z2;
    p2pack[i] = make_float4(-2.0f * x2, -2.0f * y2, -2.0f * z2, s2);
    rowmin_bits[i] = INF_BITS;
    colmin_bits[i] = INF_BITS;
}

// ---------------------------------------------------------------------------
// Kernel 2: main WMMA sweep.
// Grid (64, 2) x 256 threads (8 waves). Block (bx,by): rows = 256-row strip
// bx (wave w owns 32 rows = 2 row-tiles), columns = 8192-column half by.
// One b64 B-fragment load feeds TWO v_wmma_f32_16x16x4_f32.
// D = A x B + C gives squared distances directly. Mins accumulate on raw
// values (v_min3_num trees); clamping happens once per reduced value only.
// ---------------------------------------------------------------------------
__global__ __launch_bounds__(256)
void chamfer_wmma_kernel(const float4* __restrict__ p1pack,
                         const float4* __restrict__ p2pack,
                         const float* __restrict__ sq1arr,
                         unsigned* __restrict__ rowmin_bits,
                         unsigned* __restrict__ colmin_bits) {
    __shared__ unsigned lds_colmin[COLS_PER];  // 32KB block-local column mins

    const int tid  = threadIdx.x;
    const int lane = tid & 31;
    const int wave = tid >> 5;
    const int hi   = lane >> 4;   // half-wave: 0 => K=0,1 / rows v ; 1 => K=2,3 / rows v+8
    const int ln   = lane & 15;   // row idx (A frag) / col idx (B,C,D) in tile

    for (int k = tid; k < COLS_PER; k += 256) lds_colmin[k] = INF_BITS;
    __syncthreads();

    const int rowBase = (blockIdx.x * 8 + wave) * 32;
    const int colBase = blockIdx.y * COLS_PER;

    // A fragments (loop-invariant): lane<16 -> (x,y), lane>=16 -> (z,1)
    const v2f* p1v = (const v2f*)p1pack;
    v2f afrag0 = p1v[(rowBase + ln) * 2 + hi];
    v2f afrag1 = p1v[(rowBase + 16 + ln) * 2 + hi];

    // C fragments: C[v] = ||a_row||^2 for row = base + 8*hi + v
    v8f cfrag0, cfrag1;
#pragma unroll
    for (int v = 0; v < 8; ++v) {
        cfrag0[v] = sq1arr[rowBase + 8 * hi + v];
        cfrag1[v] = sq1arr[rowBase + 16 + 8 * hi + v];
    }

    float rmin0[8], rmin1[8];
#pragma unroll
    for (int v = 0; v < 8; ++v) { rmin0[v] = INFINITY; rmin1[v] = INFINITY; }

    const v2f* p2v = (const v2f*)p2pack;

    // software pipeline: prefetch next B fragment while WMMAs consume current
    v2f bnext = p2v[(colBase + ln) * 2 + hi];

#pragma unroll 2
    for (int ct = 0; ct < TILES_PER; ++ct) {
        v2f bcur = bnext;
        int nxt = (ct + 1 < TILES_PER) ? (ct + 1) : 0;   // tail prefetch wraps
        bnext = p2v[(colBase + nxt * 16 + ln) * 2 + hi];

        // D[m][n] = ||a_m||^2 + ||b_n||^2 - 2 a_m.b_n  (one instruction each)
        v8f d0 = __builtin_amdgcn_wmma_f32_16x16x4_f32(
            false, afrag0, false, bcur, (short)0, cfrag0, false, false);
        v8f d1 = __builtin_amdgcn_wmma_f32_16x16x4_f32(
            false, afrag1, false, bcur, (short)0, cfrag1, false, false);

        float cm = INFINITY;
#pragma unroll
        for (int v = 0; v < 8; ++v) {
            rmin0[v] = fminf(rmin0[v], d0[v]);
            rmin1[v] = fminf(rmin1[v], d1[v]);
            cm = fminf(cm, fminf(d0[v], d1[v]));
        }
        // Clamp ONCE per tile, then branchless column min: lanes ln and ln+16
        // (rows 0-7 / 8-15) hit the same address; DS serializes the conflict.
        atomicMin(&lds_colmin[ct * 16 + ln], __float_as_uint(fmaxf(cm, 0.0f)));
    }

    // Row mins: reduce across the 16 lanes of each half (masks stay in-half),
    // clamp once, then merge the two column-halves via global atomic min.
#pragma unroll
    for (int v = 0; v < 8; ++v) {
        float r0 = rmin0[v], r1 = rmin1[v];
        r0 = fminf(r0, __shfl_xor(r0, 1, 32));
        r0 = fminf(r0, __shfl_xor(r0, 2, 32));
        r0 = fminf(r0, __shfl_xor(r0, 4, 32));
        r0 = fminf(r0, __shfl_xor(r0, 8, 32));
        r1 = fminf(r1, __shfl_xor(r1, 1, 32));
        r1 = fminf(r1, __shfl_xor(r1, 2, 32));
        r1 = fminf(r1, __shfl_xor(r1, 4, 32));
        r1 = fminf(r1, __shfl_xor(r1, 8, 32));
        if (ln == 0) {
            atomicMin(&rowmin_bits[rowBase + 8 * hi + v],
                      __float_as_uint(fmaxf(r0, 0.0f)));
            atomicMin(&rowmin_bits[rowBase + 16 + 8 * hi + v],
                      __float_as_uint(fmaxf(r1, 0.0f)));
        }
    }

    __syncthreads();
    // Flush block-local column mins into the global table (branchless).
    for (int k = tid; k < COLS_PER; k += 256)
        atomicMin(&colmin_bits[colBase + k], lds_colmin[k]);
}

// ---------------------------------------------------------------------------
// Kernel 3: out = sum sqrt(rowmin) + sum sqrt(colmin)   (values pre-clamped)
// ---------------------------------------------------------------------------
__global__ __launch_bounds__(256)
void chamfer_reduce_kernel(const unsigned* __restrict__ rowmin_bits,
                           const unsigned* __restrict__ colmin_bits,
                           float* __restrict__ out) {
    __shared__ float sbuf[256];
    float acc = 0.0f;
    for (int k = threadIdx.x; k < NPTS; k += 256) {
        acc += sqrtf(__uint_as_float(rowmin_bits[k]));
        acc += sqrtf(__uint_as_float(colmin_bits[k]));
    }
    sbuf[threadIdx.x] = acc;
    __syncthreads();
    for (int s = 128; s > 0; s >>= 1) {
        if (threadIdx.x < s) sbuf[threadIdx.x] += sbuf[threadIdx.x + s];
        __syncthreads();
    }
    if (threadIdx.x == 0) out[0] = sbuf[0];
}

// ---------------------------------------------------------------------------
extern "C" void kernel_launch(void* const* d_in, const int* in_sizes, int n_in,
                              void* d_out, int out_size, void* d_ws, size_t ws_size,
                              hipStream_t stream) {
    const float* p1 = (const float*)d_in[0];   // [16384,3] f32
    const float* p2 = (const float*)d_in[1];   // [16384,3] f32
    float* out = (float*)d_out;                // scalar f32

    char* ws = (char*)d_ws;
    float4*   p1pack = (float4*)(ws);                                    // 256 KB
    float4*   p2pack = (float4*)(ws + (size_t)NPTS * 16);                // 256 KB
    float*    sq1arr = (float*)   (ws + (size_t)NPTS * 32);              //  64 KB
    unsigned* rowmin = (unsigned*)(ws + (size_t)NPTS * 32 + NPTS * 4);   //  64 KB
    unsigned* colmin = (unsigned*)(ws + (size_t)NPTS * 32 + NPTS * 8);   //  64 KB

    chamfer_prep_kernel<<<NPTS / 256, 256, 0, stream>>>(p1, p2, p1pack, p2pack,
                                                        sq1arr, rowmin, colmin);
    dim3 grid(NPTS / (8 * 32), CSPLIT);
    chamfer_wmma_kernel<<<grid, 256, 0, stream>>>(p1pack, p2pack, sq1arr,
                                                  rowmin, colmin);
    chamfer_reduce_kernel<<<1, 256, 0, stream>>>(rowmin, colmin, out);
}